// MultiBoxLoss_combined_58042188038312
// MI455X (gfx1250) — compile-verified
//
#include <hip/hip_runtime.h>

constexpr int BB = 32;       // batch
constexpr int PPR = 16384;   // priors
constexpr int CC = 80;       // classes
constexpr int TT = 50;       // objects
constexpr int NBLK = PPR / 256;           // 64 blocks per batch
constexpr int NPART = BB * NBLK;          // 2048 partials
constexpr int RBITS = 12;                 // radix digit width
constexpr int RBINS = 1 << RBITS;         // 4096 bins
constexpr int NPASS = 4;                  // 4 x 12 = 48 bits >= 46-bit keys

typedef float v2f __attribute__((ext_vector_type(2)));
typedef float v8f __attribute__((ext_vector_type(8)));

static __device__ inline unsigned long long u64_shflxor(unsigned long long v, int m) {
    unsigned lo = (unsigned)v, hi = (unsigned)(v >> 32);
    lo = __shfl_xor(lo, m, 32);
    hi = __shfl_xor(hi, m, 32);
    return ((unsigned long long)hi << 32) | lo;
}

// ---------------- K0: zero truth keys, radix histograms, prefixes ----------
__global__ void k_zero(unsigned long long* truthkey, unsigned long long* prefix,
                       unsigned* ghist) {
    int i = blockIdx.x * blockDim.x + threadIdx.x;
    int n = NPASS * 2 * BB * RBINS;
    if (i < n) ghist[i] = 0u;
    if (i < BB * TT) truthkey[i] = 0ull;
    if (i < 2 * BB) prefix[i] = 0ull;
}

// ---------------- K1: IoU match ----------------
// per prior: best truth (first-max). per truth: best prior via u64 atomicMax
// key = (iou_bits << 32) | ~p   (iou >= 0 -> bits monotone; ties -> smallest p)
__global__ void k_match(const float* __restrict__ priors,
                        const float* __restrict__ targets,
                        float* __restrict__ bestov, int* __restrict__ bestidx,
                        unsigned long long* __restrict__ truthkey) {
    int b = blockIdx.y;
    int tid = threadIdx.x;
    int p = blockIdx.x * 256 + tid;
    __shared__ float tx1[TT], ty1[TT], tx2[TT], ty2[TT], ta[TT];
    if (tid < TT) {
        const float* tr = targets + ((size_t)b * TT + tid) * 6;
        float a = tr[0], bq = tr[1], c = tr[2], d = tr[3];
        tx1[tid] = a; ty1[tid] = bq; tx2[tid] = c; ty2[tid] = d;
        ta[tid] = (c - a) * (d - bq);
    }
    __syncthreads();
    const float* pr = priors + (size_t)p * 4;
    float cx = pr[0], cy = pr[1], pw = pr[2], ph = pr[3];
    float px1 = cx - pw * 0.5f, py1 = cy - ph * 0.5f;
    float px2 = cx + pw * 0.5f, py2 = cy + ph * 0.5f;
    float pa = (px2 - px1) * (py2 - py1);
    float bestv = -1.0f; int besti = 0;
    for (int t = 0; t < TT; ++t) {
        float ix1 = fmaxf(tx1[t], px1), iy1 = fmaxf(ty1[t], py1);
        float ix2 = fminf(tx2[t], px2), iy2 = fminf(ty2[t], py2);
        float iw = fmaxf(ix2 - ix1, 0.0f), ih = fmaxf(iy2 - iy1, 0.0f);
        float inter = iw * ih;
        float iou = inter / (ta[t] + pa - inter);
        if (iou > bestv) { bestv = iou; besti = t; }
        unsigned long long key =
            ((unsigned long long)__float_as_uint(iou) << 32) | (unsigned)(~(unsigned)p);
        for (int off = 16; off > 0; off >>= 1) {
            unsigned long long o = u64_shflxor(key, off);
            if (o > key) key = o;
        }
        if ((tid & 31) == 0) atomicMax(&truthkey[b * TT + t], key);
    }
    size_t gp = (size_t)b * PPR + p;
    bestov[gp] = bestv;
    bestidx[gp] = besti;
}

// ---------------- K2: force-match override (t ascending; later t wins) ------
__global__ void k_override(const unsigned long long* __restrict__ truthkey,
                           float* __restrict__ bestov, int* __restrict__ bestidx) {
    int b = blockIdx.x;
    if (threadIdx.x == 0) {
        for (int t = 0; t < TT; ++t) {
            unsigned long long k = truthkey[b * TT + t];
            unsigned p = ~(unsigned)(k & 0xFFFFFFFFull);
            size_t gp = (size_t)b * PPR + p;
            bestov[gp] = 2.0f;
            bestidx[gp] = t;
        }
    }
}

// ---------------- K3: encode + smooth-L1 + num_pos partials ----------------
__global__ void k_loc(const float* __restrict__ loc, const float* __restrict__ priors,
                      const float* __restrict__ targets,
                      const float* __restrict__ bestov, const int* __restrict__ bestidx,
                      int* __restrict__ tgtArr, float* __restrict__ wArr,
                      float* __restrict__ partial_l, float* __restrict__ partial_np) {
    int b = blockIdx.y;
    int tid = threadIdx.x;
    int p = blockIdx.x * 256 + tid;
    size_t gp = (size_t)b * PPR + p;
    float ov = bestov[gp];
    int idx = bestidx[gp];
    const float* tr = targets + ((size_t)b * TT + idx) * 6;
    float lab = tr[4], wt = tr[5];
    bool pos = ov >= 0.5f;
    int tg = pos ? (int)lab : 0;
    tgtArr[gp] = tg;
    wArr[gp] = wt;
    const float* pr = priors + (size_t)p * 4;
    float cx = pr[0], cy = pr[1], pw = pr[2], ph = pr[3];
    float mx1 = tr[0], my1 = tr[1], mx2 = tr[2], my2 = tr[3];
    float gcx = ((mx1 + mx2) * 0.5f - cx) / (0.1f * pw);
    float gcy = ((my1 + my2) * 0.5f - cy) / (0.1f * ph);
    float gw = logf((mx2 - mx1) / pw) / 0.2f;
    float gh = logf((my2 - my1) / ph) / 0.2f;
    const float* ld = loc + gp * 4;
    float d0 = ld[0] - gcx, d1 = ld[1] - gcy, d2 = ld[2] - gw, d3 = ld[3] - gh;
    float s = 0.0f;
    {
        float a;
        a = fabsf(d0); s += (a < 1.0f) ? 0.5f * d0 * d0 : a - 0.5f;
        a = fabsf(d1); s += (a < 1.0f) ? 0.5f * d1 * d1 : a - 0.5f;
        a = fabsf(d2); s += (a < 1.0f) ? 0.5f * d2 * d2 : a - 0.5f;
        a = fabsf(d3); s += (a < 1.0f) ? 0.5f * d3 * d3 : a - 0.5f;
    }
    float cl = pos ? s * wt : 0.0f;
    float cn = pos ? wt : 0.0f;
    __shared__ float s1[256], s2[256];
    s1[tid] = cl; s2[tid] = cn;
    __syncthreads();
    for (int st = 128; st > 0; st >>= 1) {
        if (tid < st) { s1[tid] += s1[tid + st]; s2[tid] += s2[tid + st]; }
        __syncthreads();
    }
    if (tid == 0) {
        partial_l[b * NBLK + blockIdx.x] = s1[0];
        partial_np[b * NBLK + blockIdx.x] = s2[0];
    }
}

// ---------------- K4: per-batch num_pos -> num_neg (= initial radix rank) ---
__global__ void k_npneg(const float* __restrict__ partial_np,
                        float* __restrict__ npf, int* __restrict__ remK) {
    int b = threadIdx.x;
    if (b < BB) {
        float s = 0.0f;
        for (int i = 0; i < NBLK; ++i) s += partial_np[b * NBLK + i];
        npf[b] = s;
        int np = (int)s;                       // trunc, matches astype(int32) for s>=0
        int nn = 3 * np;
        if (nn > PPR - 1) nn = PPR - 1;
        remK[b] = nn;           // conf mining
        remK[BB + b] = nn;      // obj mining
    }
}

// ---------------- K5: cross-entropies + 46-bit mining keys + pos partials --
// key46 = (f32bits(mine) << 14) | (~p & 0x3FFF): descending key order ==
// (mine desc, index asc), identical to JAX stable argsort(-mine). Keys unique.
__global__ void k_ce(const float* __restrict__ conf, const float* __restrict__ obj,
                     const int* __restrict__ tgtArr, const float* __restrict__ wArr,
                     float* __restrict__ cewc, float* __restrict__ cewo,
                     unsigned long long* __restrict__ keyc, unsigned long long* __restrict__ keyo,
                     float* __restrict__ partial_pc, float* __restrict__ partial_po) {
    int b = blockIdx.y;
    int tid = threadIdx.x;
    int p = blockIdx.x * 256 + tid;
    size_t gp = (size_t)b * PPR + p;
    const float* cr = conf + gp * CC;
    float m = -3.4e38f;
    for (int i = 0; i < CC; i += 4) {
        float4 v = *(const float4*)(cr + i);
        m = fmaxf(m, fmaxf(fmaxf(v.x, v.y), fmaxf(v.z, v.w)));
    }
    float se = 0.0f;
    for (int i = 0; i < CC; i += 4) {
        float4 v = *(const float4*)(cr + i);
        se += expf(v.x - m) + expf(v.y - m) + expf(v.z - m) + expf(v.w - m);
    }
    float lse = m + logf(se);
    const float* ob = obj + gp * 2;
    float o0 = ob[0], o1 = ob[1];
    float mo = fmaxf(o0, o1);
    float lse2 = mo + logf(expf(o0 - mo) + expf(o1 - mo));
    int tg = tgtArr[gp];
    float wt = wArr[gp];
    float cec = (tg == 0) ? (lse2 - o0) : (lse + lse2 - o1 - cr[tg - 1]);
    float ceo = lse2 - ((tg > 0) ? o1 : o0);
    cewc[gp] = cec * wt;
    cewo[gp] = ceo * wt;
    float mc = (tg > 0) ? 0.0f : cec;
    float mo2 = (tg > 0) ? 0.0f : ceo;
    unsigned long long low = (unsigned long long)((~(unsigned)p) & 0x3FFFu);
    keyc[gp] = ((unsigned long long)__float_as_uint(mc) << 14) | low;
    keyo[gp] = ((unsigned long long)__float_as_uint(mo2) << 14) | low;
    float pc = (tg > 0) ? cec * wt : 0.0f;
    float po = (tg > 0) ? ceo * wt : 0.0f;
    __shared__ float s1[256], s2[256];
    s1[tid] = pc; s2[tid] = po;
    __syncthreads();
    for (int st = 128; st > 0; st >>= 1) {
        if (tid < st) { s1[tid] += s1[tid + st]; s2[tid] += s2[tid + st]; }
        __syncthreads();
    }
    if (tid == 0) {
        partial_pc[b * NBLK + blockIdx.x] = s1[0];
        partial_po[b * NBLK + blockIdx.x] = s2[0];
    }
}

// ---------------- K6a: radix-select histogram pass -------------------------
// grid (NBLK, BB, 2): z=0 conf keys, z=1 obj keys. Integer atomics only.
__global__ void k_hist(const unsigned long long* __restrict__ keyc,
                       const unsigned long long* __restrict__ keyo,
                       const unsigned long long* __restrict__ prefix,
                       unsigned* __restrict__ ghist, int q, int shift) {
    int b = blockIdx.y;
    int which = blockIdx.z;
    int tid = threadIdx.x;
    int p = blockIdx.x * 256 + tid;
    __shared__ unsigned hist[RBINS];
    for (int i = tid; i < RBINS; i += 256) hist[i] = 0u;
    __syncthreads();
    const unsigned long long* keys = which ? keyo : keyc;
    unsigned long long key = keys[(size_t)b * PPR + p];
    unsigned long long pref = prefix[which * BB + b];
    if (((key ^ pref) >> (shift + RBITS)) == 0ull)
        atomicAdd(&hist[(unsigned)((key >> shift) & (unsigned long long)(RBINS - 1))], 1u);
    __syncthreads();
    unsigned* gh = ghist + (((size_t)q * 2 + which) * BB + b) * RBINS;
    for (int i = tid; i < RBINS; i += 256) {
        unsigned c = hist[i];
        if (c) atomicAdd(&gh[i], c);
    }
}

// ---------------- K6b: pick digit of K-th largest key ----------------------
__global__ void k_sel(const unsigned* __restrict__ ghist,
                      unsigned long long* __restrict__ prefix,
                      int* __restrict__ remK, int q, int shift) {
    int i = blockIdx.x * blockDim.x + threadIdx.x;   // 0..2*BB-1
    if (i >= 2 * BB) return;
    int rem = remK[i];
    if (rem <= 0) { prefix[i] = ~0ull; return; }     // K==0: select nothing
    const unsigned* gh = ghist + ((size_t)q * 2 * BB + i) * RBINS;
    unsigned cum = 0;
    for (int bin = RBINS - 1; bin >= 0; --bin) {
        unsigned c = gh[bin];
        if (cum + c >= (unsigned)rem) {
            prefix[i] |= ((unsigned long long)bin << shift);
            remK[i] = rem - (int)cum;
            return;
        }
        cum += c;
    }
    prefix[i] = ~0ull;   // safety (unreachable)
}

// ---------------- K6c: selected-negative sums (key >= T) -------------------
__global__ void k_negsum(const unsigned long long* __restrict__ keyc,
                         const unsigned long long* __restrict__ keyo,
                         const float* __restrict__ cewc, const float* __restrict__ cewo,
                         const int* __restrict__ tgtArr,
                         const unsigned long long* __restrict__ prefix,
                         float* __restrict__ partial_nc, float* __restrict__ partial_no) {
    int b = blockIdx.y;
    int tid = threadIdx.x;
    int p = blockIdx.x * 256 + tid;
    size_t gp = (size_t)b * PPR + p;
    unsigned long long Tc = prefix[b];
    unsigned long long To = prefix[BB + b];
    int tg = tgtArr[gp];
    float vc = (tg == 0 && keyc[gp] >= Tc) ? cewc[gp] : 0.0f;
    float vo = (tg == 0 && keyo[gp] >= To) ? cewo[gp] : 0.0f;
    __shared__ float s1[256], s2[256];
    s1[tid] = vc; s2[tid] = vo;
    __syncthreads();
    for (int st = 128; st > 0; st >>= 1) {
        if (tid < st) { s1[tid] += s1[tid + st]; s2[tid] += s2[tid + st]; }
        __syncthreads();
    }
    if (tid == 0) {
        partial_nc[b * NBLK + blockIdx.x] = s1[0];
        partial_no[b * NBLK + blockIdx.x] = s2[0];
    }
}

// ---------------- K7: final reduce via WMMA + divide by N ----------------
static __device__ inline float wmma_sum2048(const float* __restrict__ arr, int lane) {
    v8f c = {};
    v2f ones; ones[0] = 1.0f; ones[1] = 1.0f;
    for (int i = 0; i < NPART; i += 64) {
        v2f a; a[0] = arr[i + lane]; a[1] = arr[i + 32 + lane];
        // D[m][n] += sum_k A[m][k] * 1  -> layout-independent grand total
        c = __builtin_amdgcn_wmma_f32_16x16x4_f32(false, a, false, ones,
                                                  (short)0, c, false, false);
    }
    float t = c[0] + c[1] + c[2] + c[3] + c[4] + c[5] + c[6] + c[7];
    t += __shfl_xor(t, 16, 32);   // combine M=0..7 (lane j) with M=8..15 (lane j+16)
    return t;                      // valid on lanes 0..15
}

__global__ void k_final(const float* __restrict__ partial_l,
                        const float* __restrict__ partial_pc,
                        const float* __restrict__ partial_po,
                        const float* __restrict__ partial_nc,
                        const float* __restrict__ partial_no,
                        const float* __restrict__ npf,
                        float* __restrict__ out) {
    int lane = threadIdx.x;  // 32 threads, single wave, EXEC all ones for WMMA
    float Ll = wmma_sum2048(partial_l, lane);
    float Lc = wmma_sum2048(partial_pc, lane);
    float Lo = wmma_sum2048(partial_po, lane);
    float Nc = wmma_sum2048(partial_nc, lane);
    float No = wmma_sum2048(partial_no, lane);
    if (lane == 0) {
        float N = 0.0f;
        for (int b = 0; b < BB; ++b) N += npf[b];
        out[0] = Ll / N;
        out[1] = (Lc + Nc) / N;
        out[2] = (Lo + No) / N;
    }
}

extern "C" void kernel_launch(void* const* d_in, const int* in_sizes, int n_in,
                              void* d_out, int out_size, void* d_ws, size_t ws_size,
                              hipStream_t stream) {
    (void)in_sizes; (void)n_in; (void)out_size; (void)ws_size;
    const float* loc     = (const float*)d_in[0];
    const float* conf    = (const float*)d_in[1];
    const float* obj     = (const float*)d_in[2];
    const float* priors  = (const float*)d_in[3];
    const float* targets = (const float*)d_in[4];
    float* out = (float*)d_out;

    char* base = (char*)d_ws;
    size_t off = 0;
    auto alloc = [&](size_t bytes) -> void* {
        off = (off + 255) & ~(size_t)255;
        void* p = base + off;
        off += bytes;
        return p;
    };
    const size_t BP = (size_t)BB * PPR;
    float*              bestov    = (float*)alloc(BP * 4);
    int*                bestidx   = (int*)alloc(BP * 4);
    int*                tgtArr    = (int*)alloc(BP * 4);
    float*              wArr      = (float*)alloc(BP * 4);
    float*              cewc      = (float*)alloc(BP * 4);
    float*              cewo      = (float*)alloc(BP * 4);
    unsigned long long* keyc      = (unsigned long long*)alloc(BP * 8);
    unsigned long long* keyo      = (unsigned long long*)alloc(BP * 8);
    unsigned long long* truthkey  = (unsigned long long*)alloc((size_t)BB * TT * 8);
    unsigned*           ghist     = (unsigned*)alloc((size_t)NPASS * 2 * BB * RBINS * 4);
    unsigned long long* prefix    = (unsigned long long*)alloc(2 * BB * 8);
    int*                remK      = (int*)alloc(2 * BB * 4);
    float*              partial_l = (float*)alloc(NPART * 4);
    float*              partial_np= (float*)alloc(NPART * 4);
    float*              partial_pc= (float*)alloc(NPART * 4);
    float*              partial_po= (float*)alloc(NPART * 4);
    float*              partial_nc= (float*)alloc(NPART * 4);
    float*              partial_no= (float*)alloc(NPART * 4);
    float*              npf       = (float*)alloc(BB * 4);

    int nz = NPASS * 2 * BB * RBINS;
    k_zero<<<(nz + 255) / 256, 256, 0, stream>>>(truthkey, prefix, ghist);
    k_match<<<dim3(NBLK, BB), 256, 0, stream>>>(priors, targets, bestov, bestidx, truthkey);
    k_override<<<BB, 32, 0, stream>>>(truthkey, bestov, bestidx);
    k_loc<<<dim3(NBLK, BB), 256, 0, stream>>>(loc, priors, targets, bestov, bestidx,
                                              tgtArr, wArr, partial_l, partial_np);
    k_npneg<<<1, 32, 0, stream>>>(partial_np, npf, remK);
    k_ce<<<dim3(NBLK, BB), 256, 0, stream>>>(conf, obj, tgtArr, wArr, cewc, cewo,
                                             keyc, keyo, partial_pc, partial_po);
    for (int q = 0; q < NPASS; ++q) {
        int shift = 36 - 12 * q;
        k_hist<<<dim3(NBLK, BB, 2), 256, 0, stream>>>(keyc, keyo, prefix, ghist, q, shift);
        k_sel<<<1, 64, 0, stream>>>(ghist, prefix, remK, q, shift);
    }
    k_negsum<<<dim3(NBLK, BB), 256, 0, stream>>>(keyc, keyo, cewc, cewo, tgtArr,
                                                 prefix, partial_nc, partial_no);
    k_final<<<1, 32, 0, stream>>>(partial_l, partial_pc, partial_po,
                                  partial_nc, partial_no, npf, out);
}